// ALiBiAttention_9010841387161
// MI455X (gfx1250) — compile-verified
//
#include <hip/hip_runtime.h>
#include <math.h>
#include <stdint.h>

typedef float v2f __attribute__((ext_vector_type(2)));
typedef float v8f __attribute__((ext_vector_type(8)));

#define B_ 2
#define T_ 2048
#define C_ 1024
#define H_ 16
#define D_ 64

// TDM availability (device pass only; host pass parses fallback path)
#ifndef __has_builtin
#define __has_builtin(x) 0
#endif
#if defined(__HIP_DEVICE_COMPILE__) && \
    __has_builtin(__builtin_amdgcn_tensor_load_to_lds) && \
    __has_builtin(__builtin_amdgcn_s_wait_tensorcnt)
#define USE_TDM 1
#else
#define USE_TDM 0
#endif

__device__ __forceinline__ v8f wmma_f32(v2f a, v2f b, v8f c) {
  // V_WMMA_F32_16X16X4_F32 : D = A(16x4) * B(4x16) + C(16x16), fp32
  return __builtin_amdgcn_wmma_f32_16x16x4_f32(false, a, false, b, (short)0, c,
                                               false, false);
}

#if USE_TDM
typedef unsigned int v4u __attribute__((ext_vector_type(4)));
typedef int v4i __attribute__((ext_vector_type(4)));
typedef int v8i __attribute__((ext_vector_type(8)));

// DMA a 16x64 fp32 tile (rows contiguous, row stride 64 floats in memory)
// into LDS with 4-dword padding after each 64-dword row => LDS stride 68.
__device__ __forceinline__ void tdm_load_tile_16x64(const float* gsrc,
                                                    unsigned lds_addr) {
  uint64_t ga = (uint64_t)(uintptr_t)gsrc;
  v4u g0;
  g0[0] = 1u;                                    // count=1, user descriptor
  g0[1] = lds_addr;                              // LDS byte address
  g0[2] = (unsigned)(ga & 0xffffffffu);          // global_addr[31:0]
  g0[3] = (unsigned)((ga >> 32) & 0x01ffffffu)   // global_addr[56:32]
          | (2u << 30);                          // type = 2 ("image")
  v8i g1;
  g1[0] = (int)((2u << 16)      // data_size = 4B
                | (1u << 20)    // pad_enable
                | (5u << 22)    // pad_interval code 5 -> every 64 dwords
                | (3u << 25));  // pad_amount  code 3 -> 4 dwords
  g1[1] = (int)(64u << 16);     // tensor_dim0 = 64 (bits 79:48 low half)
  g1[2] = (int)(16u << 16);     // tensor_dim1 = 16 (bits 111:80 low half)
  g1[3] = (int)(64u << 16);     // tile_dim0 = 64 (bits 127:112)
  g1[4] = 16;                   // tile_dim1 = 16 (bits 143:128)
  g1[5] = 64;                   // tensor_dim0_stride = 64 (bits 207:160)
  g1[6] = 0;
  g1[7] = 0;
  v4i zz = {0, 0, 0, 0};
#if __clang_major__ >= 23
  v8i zz8 = {0, 0, 0, 0, 0, 0, 0, 0};
  __builtin_amdgcn_tensor_load_to_lds(g0, g1, zz, zz, zz8, 0);
#else
  __builtin_amdgcn_tensor_load_to_lds(g0, g1, zz, zz, 0);
#endif
}
#endif  // USE_TDM

// ---------------- QKV projection: y = x @ W^T, scatter to (B,H,T,D) ----------
// 32(M) x 64(N) tile per wave: two A-fragments share four B-fragments.
__global__ __launch_bounds__(32) void qkv_proj_kernel(
    const float* __restrict__ x,
    const float* __restrict__ Wq, const float* __restrict__ Wk,
    const float* __restrict__ Wv,
    float* __restrict__ qb, float* __restrict__ kb, float* __restrict__ vb)
{
  const int lane = threadIdx.x;
  const int half = lane >> 4, l16 = lane & 15;
  const int mBase = blockIdx.x * 32;
  const int nBase = blockIdx.y * 64;
  const float* W; float* out;
  if (blockIdx.z == 0)      { W = Wq; out = qb; }
  else if (blockIdx.z == 1) { W = Wk; out = kb; }
  else                      { W = Wv; out = vb; }

  v8f acc0[4], acc1[4];
  #pragma unroll
  for (int j = 0; j < 4; ++j)
    #pragma unroll
    for (int r = 0; r < 8; ++r) { acc0[j][r] = 0.0f; acc1[j][r] = 0.0f; }

  const float* xrow0 = x + (size_t)(mBase + l16) * C_;
  const float* xrow1 = xrow0 + (size_t)16 * C_;
  const float* wrow  = W + (size_t)(nBase + l16) * C_;
  for (int k = 0; k < C_; k += 4) {
    v2f a0 = *(const v2f*)(xrow0 + k + 2 * half);
    v2f a1 = *(const v2f*)(xrow1 + k + 2 * half);
    #pragma unroll
    for (int j = 0; j < 4; ++j) {
      v2f b = *(const v2f*)(wrow + (size_t)j * 16 * C_ + k + 2 * half);
      acc0[j] = wmma_f32(a0, b, acc0[j]);
      acc1[j] = wmma_f32(a1, b, acc1[j]);
    }
  }

  #pragma unroll
  for (int j = 0; j < 4; ++j) {
    #pragma unroll
    for (int r = 0; r < 8; ++r) {
      int n = nBase + j * 16 + l16;           // h*D + d
      int h = n >> 6, d = n & (D_ - 1);
      int m0 = mBase + r + 8 * half;          // b*T + t
      int b0 = m0 >> 11, t0 = m0 & (T_ - 1);
      out[((size_t)(b0 * H_ + h) * T_ + t0) * D_ + d] = acc0[j][r];
      int m1 = m0 + 16;
      int b1 = m1 >> 11, t1 = m1 & (T_ - 1);
      out[((size_t)(b1 * H_ + h) * T_ + t1) * D_ + d] = acc1[j][r];
    }
  }
}

// ---------------- Output projection: Y = X @ Wo^T (row-major) ----------------
__global__ __launch_bounds__(32) void out_proj_kernel(
    const float* __restrict__ X, const float* __restrict__ Wo,
    float* __restrict__ Y)
{
  const int lane = threadIdx.x;
  const int half = lane >> 4, l16 = lane & 15;
  const int mBase = blockIdx.x * 32;
  const int nBase = blockIdx.y * 64;

  v8f acc0[4], acc1[4];
  #pragma unroll
  for (int j = 0; j < 4; ++j)
    #pragma unroll
    for (int r = 0; r < 8; ++r) { acc0[j][r] = 0.0f; acc1[j][r] = 0.0f; }

  const float* xrow0 = X + (size_t)(mBase + l16) * C_;
  const float* xrow1 = xrow0 + (size_t)16 * C_;
  const float* wrow  = Wo + (size_t)(nBase + l16) * C_;
  for (int k = 0; k < C_; k += 4) {
    v2f a0 = *(const v2f*)(xrow0 + k + 2 * half);
    v2f a1 = *(const v2f*)(xrow1 + k + 2 * half);
    #pragma unroll
    for (int j = 0; j < 4; ++j) {
      v2f b = *(const v2f*)(wrow + (size_t)j * 16 * C_ + k + 2 * half);
      acc0[j] = wmma_f32(a0, b, acc0[j]);
      acc1[j] = wmma_f32(a1, b, acc1[j]);
    }
  }

  #pragma unroll
  for (int j = 0; j < 4; ++j) {
    #pragma unroll
    for (int r = 0; r < 8; ++r) {
      int n = nBase + j * 16 + l16;
      int m0 = mBase + r + 8 * half;
      Y[(size_t)m0 * C_ + n] = acc0[j][r];
      Y[(size_t)(m0 + 16) * C_ + n] = acc1[j][r];
    }
  }
}

// ---------------- Flash attention with ALiBi bias (causal) ------------------
#define KRS 68  // LDS row stride (floats) for TDM-padded K/V tiles
#define PRS 18  // LDS row stride for P transpose tile (even => 8B-aligned b64)

__global__ __launch_bounds__(32) void attn_kernel(
    const float* __restrict__ qb, const float* __restrict__ kb,
    const float* __restrict__ vb, float* __restrict__ ob)
{
  __shared__ __align__(16) float pbuf[16 * PRS];
#if USE_TDM
  __shared__ __align__(16) float ktile[16 * KRS + 4];
  __shared__ __align__(16) float vtile[16 * KRS + 4];
#endif
  const int lane = threadIdx.x;
  const int half = lane >> 4, l16 = lane & 15;
  const int qt = blockIdx.x, h = blockIdx.y, b = blockIdx.z;
  const int qBase = qt * 16;
  const size_t headoff = (size_t)(b * H_ + h) * T_ * D_;
  const float* Qp = qb + headoff;
  const float* Kp = kb + headoff;
  const float* Vp = vb + headoff;
  const float slope = exp2f(-0.5f * (float)(h + 1));  // H=16 power-of-2 slopes
  const float scale = 0.125f;                         // 1/sqrt(64)

  // preload Q tile (16x64) as 16 A-fragments
  v2f qa[16];
  const float* qrow = Qp + (size_t)(qBase + l16) * D_;
  #pragma unroll
  for (int kk = 0; kk < 16; ++kk)
    qa[kk] = *(const v2f*)(qrow + kk * 4 + 2 * half);

  v8f o[4];
  float mx[8], sm[8];
  #pragma unroll
  for (int j = 0; j < 4; ++j)
    #pragma unroll
    for (int r = 0; r < 8; ++r) o[j][r] = 0.0f;
  #pragma unroll
  for (int r = 0; r < 8; ++r) { mx[r] = -1e30f; sm[r] = 0.0f; }

  for (int kt = 0; kt <= qt; ++kt) {
    const int kBase = kt * 16;

#if USE_TDM
    // Async DMA of K and V 16x64 tiles into padded LDS (TENSORcnt-tracked).
    tdm_load_tile_16x64(Kp + (size_t)kBase * D_,
                        (unsigned)(uintptr_t)&ktile[0]);
    tdm_load_tile_16x64(Vp + (size_t)kBase * D_,
                        (unsigned)(uintptr_t)&vtile[0]);
    __builtin_amdgcn_s_wait_tensorcnt(0);
#endif

    // S = Q @ K^T  (16x16)
    v8f s;
    #pragma unroll
    for (int r = 0; r < 8; ++r) s[r] = 0.0f;
    #pragma unroll
    for (int kk = 0; kk < 16; ++kk) {
#if USE_TDM
      v2f kf = *(const v2f*)(&ktile[l16 * KRS + kk * 4 + 2 * half]);
#else
      v2f kf = *(const v2f*)(Kp + (size_t)(kBase + l16) * D_ + kk * 4 + 2 * half);
#endif
      s = wmma_f32(qa[kk], kf, s);
    }

    // bias + causal mask + online softmax (row = r + 8*half, col = l16)
    float p[8];
    #pragma unroll
    for (int r = 0; r < 8; ++r) {
      int qi = qBase + r + 8 * half;
      int ki = kBase + l16;
      float val = s[r] * scale - slope * fabsf((float)(qi - ki));
      if (ki > qi) val = -1e30f;
      float vmax = val;
      #pragma unroll
      for (int msk = 1; msk <= 8; msk <<= 1)
        vmax = fmaxf(vmax, __shfl_xor(vmax, msk, 32));
      float nm   = fmaxf(mx[r], vmax);
      float corr = __expf(mx[r] - nm);
      float pe   = __expf(val - nm);
      float psum = pe;
      #pragma unroll
      for (int msk = 1; msk <= 8; msk <<= 1)
        psum += __shfl_xor(psum, msk, 32);
      sm[r] = sm[r] * corr + psum;
      mx[r] = nm;
      #pragma unroll
      for (int j = 0; j < 4; ++j) o[j][r] *= corr;
      p[r] = pe;
    }

    // D-layout -> A-layout conversion of P through LDS
    #pragma unroll
    for (int r = 0; r < 8; ++r) pbuf[(r + 8 * half) * PRS + l16] = p[r];
    __syncthreads();

    // O += P @ V  (16x16 * 16x64)
    #pragma unroll
    for (int kk = 0; kk < 4; ++kk) {
      v2f pa = *(const v2f*)(&pbuf[l16 * PRS + kk * 4 + 2 * half]);
      const int kl = kk * 4 + 2 * half;  // key-local row
      #pragma unroll
      for (int j = 0; j < 4; ++j) {
        v2f vf;
#if USE_TDM
        vf[0] = vtile[kl * KRS + j * 16 + l16];
        vf[1] = vtile[(kl + 1) * KRS + j * 16 + l16];
#else
        vf[0] = Vp[(size_t)(kBase + kl) * D_ + j * 16 + l16];
        vf[1] = Vp[(size_t)(kBase + kl + 1) * D_ + j * 16 + l16];
#endif
        o[j] = wmma_f32(pa, vf, o[j]);
      }
    }
    __syncthreads();
  }

  // normalize and store to (B,T,C)
  #pragma unroll
  for (int j = 0; j < 4; ++j) {
    #pragma unroll
    for (int r = 0; r < 8; ++r) {
      int m = qBase + r + 8 * half;
      int d = j * 16 + l16;
      ob[(size_t)(b * T_ + m) * C_ + h * D_ + d] = o[j][r] / sm[r];
    }
  }
}

extern "C" void kernel_launch(void* const* d_in, const int* in_sizes, int n_in,
                              void* d_out, int out_size, void* d_ws, size_t ws_size,
                              hipStream_t stream) {
  (void)in_sizes; (void)n_in; (void)out_size; (void)ws_size;
  const float* x  = (const float*)d_in[0];
  const float* Wq = (const float*)d_in[1];
  const float* Wk = (const float*)d_in[2];
  const float* Wv = (const float*)d_in[3];
  const float* Wo = (const float*)d_in[4];
  float* out = (float*)d_out;

  const size_t per = (size_t)B_ * H_ * T_ * D_;  // 4M floats per buffer
  float* qb = (float*)d_ws;
  float* kb = qb + per;
  float* vb = kb + per;
  float* ab = vb + per;  // attention output in (B,T,C)

  dim3 blk(32, 1, 1);
  qkv_proj_kernel<<<dim3((B_ * T_) / 32, C_ / 64, 3), blk, 0, stream>>>(
      x, Wq, Wk, Wv, qb, kb, vb);
  attn_kernel<<<dim3(T_ / 16, H_, B_), blk, 0, stream>>>(qb, kb, vb, ab);
  out_proj_kernel<<<dim3((B_ * T_) / 32, C_ / 64, 1), blk, 0, stream>>>(
      ab, Wo, out);
}